// Net_sub_attack_2_68204080660483
// MI455X (gfx1250) — compile-verified
//
#include <hip/hip_runtime.h>
#include <hip/hip_bf16.h>

typedef float v2f __attribute__((ext_vector_type(2)));
typedef float v8f __attribute__((ext_vector_type(8)));

#define NN 10000      // nodes
#define NE 320000     // edges
#define HID 16
#define LAB 7

// ---------------- workspace layout (floats) ----------------
// [0:10000)          outdeg  -> becomes norm_src
// [10000:20000)      indeg   -> becomes norm_dst
// [20000:180000)     H1raw   (N x 16)  = (feat @ W1) * norm_src
// [180000:340000)    agg1    (N x 16)  zeroed, atomic target
// [340000:410000)    h2      (N x 7)
// [410000:480000)    agg2    (N x 7)   zeroed, atomic target

__global__ void k_zero(float* __restrict__ p, int n) {
    int i = blockIdx.x * blockDim.x + threadIdx.x;
    if (i < n) p[i] = 0.0f;
}

__global__ void k_degrees(const int* __restrict__ src, const int* __restrict__ dst,
                          float* __restrict__ outdeg, float* __restrict__ indeg) {
    int e = blockIdx.x * blockDim.x + threadIdx.x;
    if (e < NE) {
        atomicAdd(&outdeg[src[e]], 1.0f);
        atomicAdd(&indeg[dst[e]], 1.0f);
    }
}

__global__ void k_norms(float* __restrict__ outdeg, float* __restrict__ indeg) {
    int n = blockIdx.x * blockDim.x + threadIdx.x;
    if (n < NN) {
        outdeg[n] = rsqrtf(fmaxf(outdeg[n], 1.0f));
        indeg[n]  = rsqrtf(fmaxf(indeg[n],  1.0f));
    }
}

// ---------------- Layer-1 GEMM: H1 = (feat @ W1) * norm_src ----------------
// One block = one 16-row tile, 8 waves. K=10000 split as 625 groups of 16
// columns; wave w takes groups w, w+8, ... so the block streams 512B/row/step.
// Each group = 4 chained V_WMMA_F32_16X16X4_F32 ops; norm_src is applied at
// writeback (it commutes through the GEMM), keeping the inner loop load+WMMA.
__global__ __launch_bounds__(256) void k_gemm1_wmma(
    const float* __restrict__ feat, const float* __restrict__ W1,
    const float* __restrict__ norm_src, float* __restrict__ H1)
{
    __shared__ float red[8 * 256];
    const int lane = threadIdx.x & 31;
    const int wave = threadIdx.x >> 5;
    const int r0   = blockIdx.x * 16;
    const int row  = r0 + (lane & 15);
    const int khalf = (lane >> 4) << 1;        // 0 for lanes 0-15, 2 for 16-31
    const int ncol  = lane & 15;               // B column (N index)
    const float* __restrict__ arow = feat + (size_t)row * NN;

    v8f acc = {0.f, 0.f, 0.f, 0.f, 0.f, 0.f, 0.f, 0.f};

    const int ngroups = NN / 16;               // 625 groups of 4 chunks
    for (int g = wave; g < ngroups; g += 8) {
        const int kbase = g * 16;
        // stream ahead: each lane prefetches its own row 2KB ahead
        __builtin_prefetch(arow + kbase + 512, 0, 0);
#pragma unroll
        for (int u = 0; u < 4; ++u) {
            const int kc = kbase + u * 4;
            // A fragment (ISA 16x4 f32 layout): VGPR0 = K even, VGPR1 = K odd
            v2f a = *(const v2f*)(arow + kc + khalf);
            // B fragment: rows kc+khalf, kc+khalf+1 of W1, column = lane&15
            v2f b;
            b.x = W1[(kc + khalf)     * HID + ncol];
            b.y = W1[(kc + khalf + 1) * HID + ncol];
            acc = __builtin_amdgcn_wmma_f32_16x16x4_f32(
                      false, a, false, b, (short)0, acc, false, false);
        }
    }

    // cross-wave reduction of the 16x16 f32 tile
    *(v8f*)(&red[wave * 256 + lane * 8]) = acc;
    __syncthreads();

    const int t = threadIdx.x;                 // 0..255 -> (lane l, vreg v)
    float s = 0.f;
#pragma unroll
    for (int w = 0; w < 8; ++w) s += red[w * 256 + t];

    const int l = t >> 3;                      // original lane 0..31
    const int v = t & 7;                       // accumulator VGPR 0..7
    const int M = v + ((l >> 4) << 3);         // C/D layout: M = v (+8 for hi half)
    const int Nc = l & 15;                     // N = lane % 16
    s *= norm_src[r0 + M];                     // fold D_out^{-1/2} here
    H1[(size_t)(r0 + M) * HID + Nc] = s;
}

// ---------------- edge scatter: agg1[dst] += H1[src], per (edge, j) ----------
__global__ void k_scatter16(const int* __restrict__ src, const int* __restrict__ dst,
                            const float* __restrict__ H1, float* __restrict__ agg1) {
    int t = blockIdx.x * blockDim.x + threadIdx.x;
    if (t < NE * HID) {
        int e = t >> 4;
        int j = t & 15;
        atomicAdd(&agg1[(size_t)dst[e] * HID + j], H1[(size_t)src[e] * HID + j]);
    }
}

// ------------- relu + norm + tiny GEMM2: h2 = (relu(agg1*nd+b1)*ns) @ W2 -----
__global__ void k_relu_gemm2(const float* __restrict__ agg1,
                             const float* __restrict__ norm_src,
                             const float* __restrict__ norm_dst,
                             const float* __restrict__ b1,
                             const float* __restrict__ W2,
                             float* __restrict__ h2) {
    __shared__ float sW2[HID * LAB];
    if (threadIdx.x < HID * LAB) sW2[threadIdx.x] = W2[threadIdx.x];
    __syncthreads();
    int n = blockIdx.x * blockDim.x + threadIdx.x;
    if (n >= NN) return;
    const float nd = norm_dst[n];
    const float ns = norm_src[n];
    float h[HID];
#pragma unroll
    for (int j = 0; j < HID; ++j)
        h[j] = fmaxf(agg1[(size_t)n * HID + j] * nd + b1[j], 0.0f) * ns;
#pragma unroll
    for (int c = 0; c < LAB; ++c) {
        float s = 0.f;
#pragma unroll
        for (int j = 0; j < HID; ++j) s += h[j] * sW2[j * LAB + c];
        h2[(size_t)n * LAB + c] = s;
    }
}

// ---------------- edge scatter 2: agg2[dst] += h2[src], per (edge, c) --------
__global__ void k_scatter7(const int* __restrict__ src, const int* __restrict__ dst,
                           const float* __restrict__ h2, float* __restrict__ agg2) {
    int t = blockIdx.x * blockDim.x + threadIdx.x;
    if (t < NE * LAB) {
        int e = t / LAB;
        int c = t % LAB;
        atomicAdd(&agg2[(size_t)dst[e] * LAB + c], h2[(size_t)src[e] * LAB + c]);
    }
}

// ---------------- final: out = agg2 * norm_dst + b2 --------------------------
__global__ void k_final(const float* __restrict__ agg2, const float* __restrict__ norm_dst,
                        const float* __restrict__ b2, float* __restrict__ out) {
    int t = blockIdx.x * blockDim.x + threadIdx.x;
    if (t < NN * LAB) {
        int n = t / LAB;
        int c = t % LAB;
        out[t] = agg2[t] * norm_dst[n] + b2[c];
    }
}

extern "C" void kernel_launch(void* const* d_in, const int* in_sizes, int n_in,
                              void* d_out, int out_size, void* d_ws, size_t ws_size,
                              hipStream_t stream) {
    const float* feat = (const float*)d_in[0];
    const int*   src  = (const int*)d_in[1];
    const int*   dst  = (const int*)d_in[2];
    const float* W1   = (const float*)d_in[3];
    const float* b1   = (const float*)d_in[4];
    const float* W2   = (const float*)d_in[5];
    const float* b2   = (const float*)d_in[6];
    float* out = (float*)d_out;

    float* ws      = (float*)d_ws;
    float* outdeg  = ws;                 // -> norm_src
    float* indeg   = ws + 10000;         // -> norm_dst
    float* H1      = ws + 20000;
    float* agg1    = ws + 180000;
    float* h2      = ws + 340000;
    float* agg2    = ws + 410000;

    const int B = 256;
    // zero everything we accumulate into (deterministic across graph replays)
    k_zero<<<(480000 + B - 1) / B, B, 0, stream>>>(ws, 480000);
    // degrees + norms
    k_degrees<<<(NE + B - 1) / B, B, 0, stream>>>(src, dst, outdeg, indeg);
    k_norms<<<(NN + B - 1) / B, B, 0, stream>>>(outdeg, indeg);
    // layer 1 big GEMM via fp32 WMMA: 625 row tiles
    k_gemm1_wmma<<<NN / 16, 256, 0, stream>>>(feat, W1, outdeg, H1);
    // layer 1 aggregation
    k_scatter16<<<(NE * HID + B - 1) / B, B, 0, stream>>>(src, dst, H1, agg1);
    // relu + norms + layer 2 dense part
    k_relu_gemm2<<<(NN + B - 1) / B, B, 0, stream>>>(agg1, outdeg, indeg, b1, W2, h2);
    // layer 2 aggregation
    k_scatter7<<<(NE * LAB + B - 1) / B, B, 0, stream>>>(src, dst, h2, agg2);
    // final bias + norm
    k_final<<<(NN * LAB + B - 1) / B, B, 0, stream>>>(agg2, indeg, b2, out);
}